// ResGCNBlock_25666724560909
// MI455X (gfx1250) — compile-verified
//
#include <hip/hip_runtime.h>
#include <math.h>

// ---------------------------------------------------------------------------
// ResGCNBlock for MI455X (gfx1250, wave32).
// Bandwidth-bound (~1.3 GB of edge gather/scatter traffic dominates at
// 23.3 TB/s); the two 100k x 128 x 128 GEMMs are done in exact fp32 with
// V_WMMA_F32_16X16X4_F32 so they are negligible and bit-accurate-ish vs the
// fp32 reference.
// ---------------------------------------------------------------------------

#define DFEAT 128

typedef __attribute__((ext_vector_type(2))) float v2f;
typedef __attribute__((ext_vector_type(8))) float v8f;

// ---------------- degree / normalization helpers ----------------

__global__ void k_fill1(float* __restrict__ deg, int n) {
    int i = blockIdx.x * blockDim.x + threadIdx.x;
    if (i < n) deg[i] = 1.0f;            // self-loop weight
}

__global__ void k_degacc(const int* __restrict__ col, const float* __restrict__ w,
                         float* __restrict__ deg, int E) {
    int e = blockIdx.x * blockDim.x + threadIdx.x;
    if (e < E) atomicAdd(&deg[col[e]], w[e]);
}

__global__ void k_dinv(float* __restrict__ deg, int n) {
    int i = blockIdx.x * blockDim.x + threadIdx.x;
    if (i < n) {
        float d = deg[i];
        deg[i] = (d > 0.0f) ? rsqrtf(d) : 0.0f;
    }
}

__global__ void k_norm(const int* __restrict__ row, const int* __restrict__ col,
                       const float* __restrict__ w, const float* __restrict__ dinv,
                       float* __restrict__ norm, int E) {
    int e = blockIdx.x * blockDim.x + threadIdx.x;
    if (e < E) norm[e] = dinv[row[e]] * w[e] * dinv[col[e]];
}

// ---------------- fp32 WMMA GEMM:  C[n x 128] = act(A) @ W[128 x 128] -------
// 256-thread block = 8 waves; each wave owns a 16-row strip and all 128
// output columns as 8 accumulator tiles; K loop steps by 4 via
// V_WMMA_F32_16X16X4_F32.
//
// W is staged in LDS as K-pair-interleaved float2 slots:
//     sW2[j*128 + nsw] = { W[2j][n], W[2j+1][n] },  nsw = (n + 16*(j&1)) & 127
// so each B operand is ONE aligned ds_load_b64 into an even VGPR pair (no
// register repacking), and the n-swizzle puts the two half-waves (which read
// adjacent j rows, 256 dwords apart == same banks) on disjoint bank halves.
//
// Per ISA layouts (wave32), h = lane>>4, lid = lane&15:
//   A 16x4 : lane holds (M=lid, K = 4kk + 2h + {0,1})
//   B 4x16 : lane holds (K = 4kk + 2h + {0,1}, N=lid) in vector elems {0,1}
//   C 16x16: vector elem r holds (M = r + 8h, N = lid)

template <bool RELU_IN>
__global__ __launch_bounds__(256) void k_gemm(const float* __restrict__ A,
                                              const float* __restrict__ W,
                                              float* __restrict__ C, int n) {
    __shared__ v2f sW2[64 * DFEAT];   // 64 KB, K-pair interleaved + swizzled

    // cooperative load + interleave + swizzle of W (one-time, coalesced b32)
    for (int i = threadIdx.x; i < 64 * DFEAT; i += 256) {
        const int j = i >> 7;          // K-pair index 0..63
        const int nn = i & 127;        // column
        v2f v;
        v[0] = W[(size_t)(2 * j) * DFEAT + nn];
        v[1] = W[(size_t)(2 * j + 1) * DFEAT + nn];
        const int nsw = (nn + ((j & 1) << 4)) & 127;
        sW2[j * DFEAT + nsw] = v;
    }
    __syncthreads();

    const int wave = threadIdx.x >> 5;
    const int lane = threadIdx.x & 31;
    const int half = lane >> 4;       // 0 or 1
    const int lid  = lane & 15;

    const int row0 = blockIdx.x * 128 + wave * 16;
    // clamp instead of predicate: rows >= n only feed C rows >= n, never stored
    const int arow = (row0 + lid < n) ? (row0 + lid) : (n - 1);
    const float* __restrict__ Arow = A + (size_t)arow * DFEAT;

    // per-lane LDS slot index (in v2f units) for each n-tile, at kk = 0
    int laddr[8];
    #pragma unroll
    for (int nt = 0; nt < 8; ++nt)
        laddr[nt] = half * DFEAT + (((nt + half) & 7) << 4) + lid;

    v8f acc[8] = {};   // 8 tiles x 8 VGPRs

    #pragma unroll 4
    for (int kk = 0; kk < 32; ++kk) {
        // A fragment: unconditional 8-byte load (cols 4kk+2h, 4kk+2h+1)
        v2f a = *reinterpret_cast<const v2f*>(Arow + kk * 4 + 2 * half);
        if (RELU_IN) { a[0] = fmaxf(a[0], 0.0f); a[1] = fmaxf(a[1], 0.0f); }

        const int kbase = kk * 2 * DFEAT;   // j = 2kk + half rows into sW2
        #pragma unroll
        for (int nt = 0; nt < 8; ++nt) {
            v2f b = sW2[kbase + laddr[nt]];   // single ds_load_b64
            acc[nt] = __builtin_amdgcn_wmma_f32_16x16x4_f32(
                false, a, false, b, (short)0, acc[nt], false, false);
        }
    }

    if (row0 + 16 <= n) {
        // fast path: whole strip in range
        #pragma unroll
        for (int nt = 0; nt < 8; ++nt) {
            const int ncol = nt * 16 + lid;
            #pragma unroll
            for (int r = 0; r < 8; ++r)
                C[(size_t)(row0 + r + 8 * half) * DFEAT + ncol] = acc[nt][r];
        }
    } else {
        #pragma unroll
        for (int nt = 0; nt < 8; ++nt) {
            const int ncol = nt * 16 + lid;
            #pragma unroll
            for (int r = 0; r < 8; ++r) {
                const int m = row0 + r + 8 * half;
                if (m < n) C[(size_t)m * DFEAT + ncol] = acc[nt][r];
            }
        }
    }
}

// ---------------- aggregation: init with bias + self-loop term --------------
// out[i,:] = b[:] + dinv[i]^2 * h[i,:]       (32 threads per node, float4)

__global__ void k_selfinit(const float* __restrict__ h, const float* __restrict__ dinv,
                           const float* __restrict__ bias, float* __restrict__ out, int n) {
    int t = blockIdx.x * blockDim.x + threadIdx.x;
    int i = t >> 5;
    if (i >= n) return;
    int f = (t & 31) * 4;
    float s = dinv[i];
    s = s * s;
    const float4 hv = *reinterpret_cast<const float4*>(h + (size_t)i * DFEAT + f);
    const float4 bv = *reinterpret_cast<const float4*>(bias + f);
    float4 o;
    o.x = bv.x + s * hv.x;
    o.y = bv.y + s * hv.y;
    o.z = bv.z + s * hv.z;
    o.w = bv.w + s * hv.w;
    *reinterpret_cast<float4*>(out + (size_t)i * DFEAT + f) = o;
}

// ---------------- aggregation: edge scatter-add -----------------------------
// 32 threads per edge, each thread handles a float4 of features.

__global__ void k_scatter(const int* __restrict__ row, const int* __restrict__ col,
                          const float* __restrict__ norm, const float* __restrict__ h,
                          float* __restrict__ out, int E) {
    int t = blockIdx.x * blockDim.x + threadIdx.x;
    int e = t >> 5;
    if (e >= E) return;
    int f = (t & 31) * 4;
    int r = row[e];
    int c = col[e];
    float nm = norm[e];
    const float4 hv = *reinterpret_cast<const float4*>(h + (size_t)r * DFEAT + f);
    float* dst = out + (size_t)c * DFEAT + f;
    atomicAdd(dst + 0, nm * hv.x);
    atomicAdd(dst + 1, nm * hv.y);
    atomicAdd(dst + 2, nm * hv.z);
    atomicAdd(dst + 3, nm * hv.w);
}

// ---------------- epilogue: relu, L2 normalize, residual --------------------
// one wave per node; lane handles 4 features; shfl-xor reduction (wave32)

__global__ __launch_bounds__(256) void k_final(const float* __restrict__ h,
                                               const float* __restrict__ x,
                                               float* __restrict__ out, int n) {
    const int wave = threadIdx.x >> 5;
    const int lane = threadIdx.x & 31;
    const int i = blockIdx.x * 8 + wave;
    if (i >= n) return;
    const int f = lane * 4;

    float4 hv = *reinterpret_cast<const float4*>(h + (size_t)i * DFEAT + f);
    hv.x = fmaxf(hv.x, 0.0f);
    hv.y = fmaxf(hv.y, 0.0f);
    hv.z = fmaxf(hv.z, 0.0f);
    hv.w = fmaxf(hv.w, 0.0f);

    float ss = hv.x * hv.x + hv.y * hv.y + hv.z * hv.z + hv.w * hv.w;
    #pragma unroll
    for (int off = 16; off > 0; off >>= 1)
        ss += __shfl_xor(ss, off, 32);

    const float nrm = sqrtf(ss);
    const float scale = 0.5f / fmaxf(nrm, 1e-12f);

    const float4 xv = *reinterpret_cast<const float4*>(x + (size_t)i * DFEAT + f);
    float4 o;
    o.x = hv.x * scale + 0.5f * xv.x;
    o.y = hv.y * scale + 0.5f * xv.y;
    o.z = hv.z * scale + 0.5f * xv.z;
    o.w = hv.w * scale + 0.5f * xv.w;
    *reinterpret_cast<float4*>(out + (size_t)i * DFEAT + f) = o;
}

// ---------------------------------------------------------------------------

extern "C" void kernel_launch(void* const* d_in, const int* in_sizes, int n_in,
                              void* d_out, int out_size, void* d_ws, size_t ws_size,
                              hipStream_t stream) {
    const float* x  = (const float*)d_in[0];
    const int*   ei = (const int*)d_in[1];
    const float* ew = (const float*)d_in[2];
    const float* W1 = (const float*)d_in[3];
    const float* b1 = (const float*)d_in[4];
    const float* W2 = (const float*)d_in[5];
    const float* b2 = (const float*)d_in[6];
    float* out = (float*)d_out;

    const int N = in_sizes[0] / DFEAT;
    const int E = in_sizes[2];
    const int* rowp = ei;        // edge_index[0]
    const int* colp = ei + E;    // edge_index[1]

    // workspace carve-out
    char* ws = (char*)d_ws;
    size_t off = 0;
    auto carve = [&](size_t bytes) -> char* {
        off = (off + 255) & ~(size_t)255;
        char* p = ws + off;
        off += bytes;
        return p;
    };
    float* dinv = (float*)carve((size_t)N * sizeof(float));
    float* norm = (float*)carve((size_t)E * sizeof(float));
    float* bufA = (float*)carve((size_t)N * DFEAT * sizeof(float)); // GEMM outputs
    float* bufB = (float*)carve((size_t)N * DFEAT * sizeof(float)); // aggregation

    const int TB = 256;
    const int gN  = (N + TB - 1) / TB;
    const int gE  = (E + TB - 1) / TB;
    const int gNf = (int)(((size_t)N * 32 + TB - 1) / TB);  // 32 thr/node
    const int gEf = (int)(((size_t)E * 32 + TB - 1) / TB);  // 32 thr/edge
    const int gG  = (N + 127) / 128;                        // GEMM blocks
    const int gFin = (N + 7) / 8;                           // 8 nodes/block

    // normalization (shared by both convs)
    k_fill1 <<<gN, TB, 0, stream>>>(dinv, N);
    k_degacc<<<gE, TB, 0, stream>>>(colp, ew, dinv, E);
    k_dinv  <<<gN, TB, 0, stream>>>(dinv, N);
    k_norm  <<<gE, TB, 0, stream>>>(rowp, colp, ew, dinv, norm, E);

    // conv1: bufA = x @ W1 ; bufB = b1 + dinv^2*bufA + scatter(norm * bufA[row])
    k_gemm<false><<<gG, TB, 0, stream>>>(x, W1, bufA, N);
    k_selfinit   <<<gNf, TB, 0, stream>>>(bufA, dinv, b1, bufB, N);
    k_scatter    <<<gEf, TB, 0, stream>>>(rowp, colp, norm, bufA, bufB, E);

    // conv2: bufA = relu(bufB) @ W2 ; bufB = b2 + dinv^2*bufA + scatter(...)
    k_gemm<true><<<gG, TB, 0, stream>>>(bufB, W2, bufA, N);
    k_selfinit  <<<gNf, TB, 0, stream>>>(bufA, dinv, b2, bufB, N);
    k_scatter   <<<gEf, TB, 0, stream>>>(rowp, colp, norm, bufA, bufB, E);

    // epilogue: relu -> L2 normalize -> (h + x)/2
    k_final<<<gFin, TB, 0, stream>>>(bufB, x, out, N);
}